// AtomConditioner_86019605004404
// MI455X (gfx1250) — compile-verified
//
#include <hip/hip_runtime.h>
#include <hip/hip_bf16.h>

typedef float v2f __attribute__((ext_vector_type(2)));
typedef float v8f __attribute__((ext_vector_type(8)));

#define C_S    256
#define C_Z    128
#define C_A    128
#define C_EMB  16
#define C_PAIR 32
#define L_RES  1024
#define N_ATOM 4096
#define C_IN   (C_S + C_EMB)   /* 272 */

#define LDSA_STRIDE 132        /* 16x128 A tile padded: conflict-free, 8B aligned */
#define XS_STRIDE   274        /* 16x272 x tile padded */
#define H1_STRIDE   132

// =====================================================================
// pair_proj: out[M=L*L, 32] = z[M,128] @ Wp[128,32] + bp
// 128 threads = 4 waves; each wave computes one 16-row x 32-col tile
// via V_WMMA_F32_16X16X4_F32, K-loop of 32 steps.
// =====================================================================
__launch_bounds__(128)
__global__ void pair_proj_kernel(const float* __restrict__ z,
                                 const float* __restrict__ Wp,
                                 const float* __restrict__ bp,
                                 float* __restrict__ out) {
  __shared__ float wpT[C_PAIR * C_Z];          // Wp transposed: [n][k], 16 KB
  __shared__ float aTile[4][16 * LDSA_STRIDE]; // per-wave A staging, ~33 KB

  const int tid  = threadIdx.x;
  const int wave = tid >> 5;
  const int lane = tid & 31;

  // Stage Wp into LDS transposed (coalesced global read).
  for (int idx = tid; idx < C_Z * C_PAIR; idx += 128) {
    const int kk = idx >> 5;   // Wp row (K)
    const int n  = idx & 31;   // Wp col (N)
    wpT[n * C_Z + kk] = Wp[idx];
  }
  __syncthreads();

  const long tile    = (long)blockIdx.x * 4 + wave;
  const long rowBase = tile * 16;

  // Stage the 16x128 fp32 A tile into LDS: each iteration all 32 lanes
  // load one full 512B row with global_load_b128 (fully coalesced).
  float* myA = &aTile[wave][0];
  {
    const float* zsrc = z + (size_t)rowBase * C_Z + lane * 4;
    #pragma unroll 4
    for (int i = 0; i < 16; ++i) {
      const float4 v = *(const float4*)(zsrc + (size_t)i * C_Z);
      *(float4*)&myA[i * LDSA_STRIDE + lane * 4] = v;
    }
  }

  const int mLane = lane & 15;          // row (A) / col (B,D) within tile
  const int kLane = (lane >> 4) * 2;    // ISA 16x4 layout: K = vgpr + 2*(lane/16)

  v8f acc0 = {};
  v8f acc1 = {};
  #pragma unroll 8
  for (int k = 0; k < C_Z; k += 4) {
    const v2f a  = *(const v2f*)&myA[mLane * LDSA_STRIDE + k + kLane];
    const v2f b0 = *(const v2f*)&wpT[mLane * C_Z + k + kLane];
    const v2f b1 = *(const v2f*)&wpT[(mLane + 16) * C_Z + k + kLane];
    acc0 = __builtin_amdgcn_wmma_f32_16x16x4_f32(false, a, false, b0,
                                                 (short)0, acc0, false, false);
    acc1 = __builtin_amdgcn_wmma_f32_16x16x4_f32(false, a, false, b1,
                                                 (short)0, acc1, false, false);
  }

  const float bb0 = bp[mLane];
  const float bb1 = bp[mLane + 16];
  float* orow = out + (size_t)rowBase * C_PAIR;
  const int mOff = (lane >> 4) * 8;     // C/D layout: VGPR r -> M = r + 8*(lane/16)
  #pragma unroll
  for (int r = 0; r < 8; ++r) {
    const int m = r + mOff;
    orow[(size_t)m * C_PAIR + mLane]      = acc0[r] + bb0;
    orow[(size_t)m * C_PAIR + 16 + mLane] = acc1[r] + bb1;
  }
}

// =====================================================================
// atom MLP: h = silu(x @ W1 + b1) @ W2 + b2, x = [s[res] || E[type]]
// 128 threads = 4 waves; block handles 16 atoms; wave w owns columns
// [32w, 32w+32). W1/W2 are read from global (L2-resident, 200 KB total).
// =====================================================================
__launch_bounds__(128)
__global__ void atom_mlp_kernel(const float* __restrict__ s,
                                const int*   __restrict__ atom_to_res,
                                const int*   __restrict__ atom_type,
                                const float* __restrict__ E_atom,
                                const float* __restrict__ W1,
                                const float* __restrict__ b1,
                                const float* __restrict__ W2,
                                const float* __restrict__ b2,
                                float* __restrict__ out) {
  __shared__ float xs[16 * XS_STRIDE];   // gathered x tile, ~17 KB
  __shared__ float h1s[16 * H1_STRIDE];  // hidden tile, ~8 KB

  const int tid  = threadIdx.x;
  const int wave = tid >> 5;
  const int lane = tid & 31;
  const int atomBase = blockIdx.x * 16;

  // Cooperative gather of the concatenated input tile.
  for (int idx = tid; idx < 16 * C_IN; idx += 128) {
    const int row = idx / C_IN;
    const int col = idx - row * C_IN;
    const int a   = atomBase + row;
    float v;
    if (col < C_S) {
      v = s[(size_t)atom_to_res[a] * C_S + col];
    } else {
      v = E_atom[atom_type[a] * C_EMB + (col - C_S)];
    }
    xs[row * XS_STRIDE + col] = v;
  }
  __syncthreads();

  const int mLane = lane & 15;
  const int kLane = (lane >> 4) * 2;
  const int mOff  = (lane >> 4) * 8;
  const int n0    = wave * 32;

  // ---- layer 1: [16,272] @ [272,128] slice, K = 272 = 68 steps ----
  v8f acc0 = {};
  v8f acc1 = {};
  #pragma unroll 4
  for (int k = 0; k < C_IN; k += 4) {
    const int kk = k + kLane;
    const v2f a = *(const v2f*)&xs[mLane * XS_STRIDE + kk];
    const v2f b0 = { W1[(size_t)kk * C_A + n0 + mLane],
                     W1[(size_t)(kk + 1) * C_A + n0 + mLane] };
    const v2f b1v = { W1[(size_t)kk * C_A + n0 + 16 + mLane],
                      W1[(size_t)(kk + 1) * C_A + n0 + 16 + mLane] };
    acc0 = __builtin_amdgcn_wmma_f32_16x16x4_f32(false, a, false, b0,
                                                 (short)0, acc0, false, false);
    acc1 = __builtin_amdgcn_wmma_f32_16x16x4_f32(false, a, false, b1v,
                                                 (short)0, acc1, false, false);
  }

  // bias + SiLU fused on accumulators, stage h1 into LDS
  {
    const float bias0 = b1[n0 + mLane];
    const float bias1 = b1[n0 + 16 + mLane];
    #pragma unroll
    for (int r = 0; r < 8; ++r) {
      float u0 = acc0[r] + bias0;
      float u1 = acc1[r] + bias1;
      u0 = u0 / (1.0f + __expf(-u0));
      u1 = u1 / (1.0f + __expf(-u1));
      h1s[(r + mOff) * H1_STRIDE + n0 + mLane]      = u0;
      h1s[(r + mOff) * H1_STRIDE + n0 + 16 + mLane] = u1;
    }
  }
  __syncthreads();

  // ---- layer 2: [16,128] @ [128,128] slice, K = 128 = 32 steps ----
  v8f c0 = {};
  v8f c1 = {};
  #pragma unroll 4
  for (int k = 0; k < C_A; k += 4) {
    const int kk = k + kLane;
    const v2f a = *(const v2f*)&h1s[mLane * H1_STRIDE + kk];
    const v2f b0 = { W2[(size_t)kk * C_A + n0 + mLane],
                     W2[(size_t)(kk + 1) * C_A + n0 + mLane] };
    const v2f b1v = { W2[(size_t)kk * C_A + n0 + 16 + mLane],
                      W2[(size_t)(kk + 1) * C_A + n0 + 16 + mLane] };
    c0 = __builtin_amdgcn_wmma_f32_16x16x4_f32(false, a, false, b0,
                                               (short)0, c0, false, false);
    c1 = __builtin_amdgcn_wmma_f32_16x16x4_f32(false, a, false, b1v,
                                               (short)0, c1, false, false);
  }

  const float bb0 = b2[n0 + mLane];
  const float bb1 = b2[n0 + 16 + mLane];
  float* orow = out + (size_t)atomBase * C_A;
  #pragma unroll
  for (int r = 0; r < 8; ++r) {
    const int m = r + mOff;
    orow[(size_t)m * C_A + n0 + mLane]      = c0[r] + bb0;
    orow[(size_t)m * C_A + n0 + 16 + mLane] = c1[r] + bb1;
  }
}

// =====================================================================
extern "C" void kernel_launch(void* const* d_in, const int* in_sizes, int n_in,
                              void* d_out, int out_size, void* d_ws, size_t ws_size,
                              hipStream_t stream) {
  const float* s     = (const float*)d_in[0];
  const float* z     = (const float*)d_in[1];
  const int*   a2r   = (const int*)d_in[2];
  const int*   atype = (const int*)d_in[3];
  const float* E     = (const float*)d_in[4];
  const float* W1    = (const float*)d_in[5];
  const float* b1    = (const float*)d_in[6];
  const float* W2    = (const float*)d_in[7];
  const float* b2    = (const float*)d_in[8];
  const float* Wp    = (const float*)d_in[9];
  const float* bp    = (const float*)d_in[10];

  float* out      = (float*)d_out;
  float* h_out    = out;                               // [4096, 128]
  float* pair_out = out + (size_t)N_ATOM * C_A;        // [L*L, 32]

  // tiny MLP first (independent); 256 blocks
  atom_mlp_kernel<<<N_ATOM / 16, 128, 0, stream>>>(
      s, a2r, atype, E, W1, b1, W2, b2, h_out);

  // dominant pair projection: 65536 row-tiles, 4 per block
  const int mTiles = (L_RES * L_RES) / 16;             // 65536
  pair_proj_kernel<<<mTiles / 4, 128, 0, stream>>>(z, Wp, bp, pair_out);
}